// SpikeEncoder_12463995093890
// MI455X (gfx1250) — compile-verified
//
#include <hip/hip_runtime.h>

#define BATCH       256
#define INPUT_DIM   512
#define NUM_NEURONS 2048
#define TIME_STEPS  100

typedef __attribute__((ext_vector_type(2))) float v2f;
typedef __attribute__((ext_vector_type(4))) float v4f;
typedef __attribute__((ext_vector_type(8))) float v8f;

// ---------------------------------------------------------------------------
// Kernel 1: rates[b][n] = sigmoid( sum_k x[b][k] * W[n][k] + bias[n] )
// Full-precision f32 GEMM on the WMMA pipe: V_WMMA_F32_16X16X4_F32.
// One 16x16 output tile per wave; K-loop of INPUT_DIM/4 = 128 steps.
//
// A fragment (16x4 f32, ISA layout): lane L holds row m = L%16,
//   K elements 2*(L/16) and 2*(L/16)+1 in its two VGPRs.
// B fragment (4x16 f32): lane L holds column n = L%16 (a row of W, since
//   B = W^T), same K-pair split across the lane halves.
// C/D (16x16 f32, 8 VGPRs): VGPR r <-> M = r + 8*(L/16), N = L%16.
// ---------------------------------------------------------------------------
__global__ __launch_bounds__(256) void rates_wmma_kernel(
    const float* __restrict__ x,      // [BATCH, INPUT_DIM]
    const float* __restrict__ W,      // [NUM_NEURONS, INPUT_DIM]
    const float* __restrict__ bias,   // [NUM_NEURONS]
    float* __restrict__ rates)        // [BATCH, NUM_NEURONS]
{
    const int lane  = threadIdx.x & 31;
    const int wave  = threadIdx.x >> 5;
    const int tile  = blockIdx.x * 8 + wave;   // 2048 tiles total
    const int tileN = tile & 127;              // 2048/16 = 128 N-tiles
    const int tileM = tile >> 7;               // 256/16  = 16  M-tiles

    const int m = lane & 15;
    const int h = lane >> 4;                   // 0: K-pair {0,1}, 1: K-pair {2,3}

    const float* xp = x + (size_t)(tileM * 16 + m) * INPUT_DIM + 2 * h;
    const float* wp = W + (size_t)(tileN * 16 + m) * INPUT_DIM + 2 * h;

    v8f acc = {};
#pragma unroll 8
    for (int k = 0; k < INPUT_DIM; k += 4) {
        v2f a  = *(const v2f*)(xp + k);
        v2f bb = *(const v2f*)(wp + k);
        // 8 args: (neg_a, A, neg_b, B, c_mod, C, reuse_a, reuse_b)
        acc = __builtin_amdgcn_wmma_f32_16x16x4_f32(
            false, a, false, bb, (short)0, acc, false, false);
    }

    const int   n     = tileN * 16 + m;
    const float bn    = bias[n];
    const int   mbase = tileM * 16 + h * 8;
#pragma unroll
    for (int r = 0; r < 8; ++r) {
        float z    = acc[r] + bn;
        float rate = 1.0f / (1.0f + __expf(-z));
        rates[(size_t)(mbase + r) * NUM_NEURONS + n] = rate;
    }
}

// ---------------------------------------------------------------------------
// Kernel 2: spikes[b][t][n] = (u[b][t][n] < rates[b][n]) ? 1.0f : 0.0f
// Pure HBM streaming (~420 MB): float4 per thread, non-temporal on the
// single-use u/spikes streams, temporal (L2-resident) loads for rates.
// Grid (2, TIME_STEPS, BATCH) x 256 threads -> no div/mod anywhere.
// ---------------------------------------------------------------------------
__global__ __launch_bounds__(256) void spikes_kernel(
    const float* __restrict__ u,      // [BATCH, TIME_STEPS, NUM_NEURONS]
    const float* __restrict__ rates,  // [BATCH, NUM_NEURONS]
    float* __restrict__ spikes)       // [BATCH, TIME_STEPS, NUM_NEURONS]
{
    const int n  = (blockIdx.x * 256 + threadIdx.x) * 4;  // neuron base
    const int t  = blockIdx.y;
    const int b  = blockIdx.z;

    const v4f r4 = *(const v4f*)(rates + (size_t)b * NUM_NEURONS + n);

    const size_t off = ((size_t)b * TIME_STEPS + t) * NUM_NEURONS + n;
    const v4f u4 = __builtin_nontemporal_load((const v4f*)(u + off));

    v4f s;
    s.x = (u4.x < r4.x) ? 1.0f : 0.0f;
    s.y = (u4.y < r4.y) ? 1.0f : 0.0f;
    s.z = (u4.z < r4.z) ? 1.0f : 0.0f;
    s.w = (u4.w < r4.w) ? 1.0f : 0.0f;

    __builtin_nontemporal_store(s, (v4f*)(spikes + off));
}

// ---------------------------------------------------------------------------
extern "C" void kernel_launch(void* const* d_in, const int* in_sizes, int n_in,
                              void* d_out, int out_size, void* d_ws, size_t ws_size,
                              hipStream_t stream)
{
    const float* x    = (const float*)d_in[0];  // [256, 512]
    const float* W    = (const float*)d_in[1];  // [2048, 512]
    const float* bias = (const float*)d_in[2];  // [2048]
    const float* u    = (const float*)d_in[3];  // [256, 100, 2048]

    float* spikes = (float*)d_out;                                      // first output
    float* rates  = spikes + (size_t)BATCH * TIME_STEPS * NUM_NEURONS;  // second output

    // 2048 16x16 tiles, 8 waves (one tile each) per 256-thread block.
    rates_wmma_kernel<<<256, 256, 0, stream>>>(x, W, bias, rates);

    dim3 grid(NUM_NEURONS / (256 * 4), TIME_STEPS, BATCH);  // (2, 100, 256)
    spikes_kernel<<<grid, 256, 0, stream>>>(u, rates, spikes);
}